// Get_line_68272800137803
// MI455X (gfx1250) — compile-verified
//
#include <hip/hip_runtime.h>
#include <stdint.h>

#define Bn   32
#define Hh   512
#define Wi   512
#define HW   (Hh * Wi)          // 262144 pixels per image
#define ROWS 16                 // stencil tile rows per block
#define LROWS (ROWS + 2)        // +2 halo rows

typedef __attribute__((ext_vector_type(4))) unsigned int v4u;
typedef __attribute__((ext_vector_type(8))) int          v8i;
typedef __attribute__((ext_vector_type(4))) int          v4i;
typedef __attribute__((ext_vector_type(4))) float        v4f;

// ---------------- Kernel 1: grayscale (float4 vectorized) ----------------
// x is streamed once -> non-temporal loads (th:NT) so it never displaces the
// L2-resident intermediates. gray is written with default (RT) policy so the
// TDM reads in kernel 2 hit L2.
__global__ __launch_bounds__(256)
void gray_kernel(const float* __restrict__ x, float* __restrict__ gray, int n4) {
    int i = blockIdx.x * blockDim.x + threadIdx.x;
    if (i >= n4) return;
    long p   = (long)i * 4;                 // flat gray index
    int  b   = (int)(p >> 18);              // / HW
    int  off = (int)(p & (HW - 1));
    const float* base = x + (size_t)b * 3 * HW + off;
    v4f r  = __builtin_nontemporal_load((const v4f*)(base));
    v4f g  = __builtin_nontemporal_load((const v4f*)(base + HW));
    v4f bl = __builtin_nontemporal_load((const v4f*)(base + 2 * HW));
    v4f o;
    o.x = 0.3f * r.x + 0.59f * g.x + 0.11f * bl.x;
    o.y = 0.3f * r.y + 0.59f * g.y + 0.11f * bl.y;
    o.z = 0.3f * r.z + 0.59f * g.z + 0.11f * bl.z;
    o.w = 0.3f * r.w + 0.59f * g.w + 0.11f * bl.w;
    *(v4f*)(gray + p) = o;
}

// ------ Kernel 2: TDM-staged 9-point stencil + deterministic reduction ------
__global__ __launch_bounds__(256)
void stencil_kernel(const float* __restrict__ gray,
                    float* __restrict__ sout,
                    float* __restrict__ partial) {
    __shared__ float tile[LROWS * Wi];      // 18 x 512 floats = 36 KB
    __shared__ float red[512];              // 2 KB reduction scratch

    const int b  = blockIdx.x >> 5;         // 32 row-tiles per image
    const int t  = blockIdx.x & 31;
    const int r0 = t * ROWS;

    const int rstart  = (r0 == 0) ? 0 : (r0 - 1);
    const int rend    = (r0 + ROWS >= Hh) ? (Hh - 1) : (r0 + ROWS);
    const int nrows   = rend - rstart + 1;
    const int ldsRow0 = (r0 == 0) ? 1 : 0;

    // Zero halo rows at image boundaries (zero padding semantics)
    if (r0 == 0) {
        for (int k = threadIdx.x; k < Wi; k += 256) tile[k] = 0.0f;
    }
    if (r0 + ROWS >= Hh) {
        for (int k = threadIdx.x; k < Wi; k += 256) tile[(LROWS - 1) * Wi + k] = 0.0f;
    }

    // ---- Tensor Data Mover: async 2D tile (Wi x nrows, 4B elems) -> LDS ----
    if (threadIdx.x < 32) {   // one wave issues the TDM descriptor
        uint32_t lds_addr = (uint32_t)(uintptr_t)(&tile[ldsRow0 * Wi]);
        uint64_t ga = (uint64_t)(uintptr_t)(gray + (size_t)b * HW + (size_t)rstart * Wi);

        v4u g0;
        g0[0] = 1u;                                   // count=1 (valid user D#)
        g0[1] = lds_addr;                             // lds_addr [63:32]
        g0[2] = (uint32_t)ga;                         // global_addr lo
        g0[3] = (uint32_t)((ga >> 32) & 0x01FFFFFFu)  // global_addr [56:32]
                | 0x80000000u;                        // type=2 ("image")

        v8i g1;
        g1[0] = 0x00020000;                           // data_size=2 (4 bytes); wg_mask=0
        g1[1] = (int)(((uint32_t)Wi & 0xFFFFu) << 16);            // tensor_dim0 lo16
        g1[2] = (int)((((uint32_t)(Hh - rstart)) & 0xFFFFu) << 16); // dim0 hi=0 | tensor_dim1 lo16
        g1[3] = (int)(((uint32_t)Wi & 0xFFFFu) << 16);            // dim1 hi=0 | tile_dim0=512
        g1[4] = nrows;                                // tile_dim1 | tile_dim2=0
        g1[5] = Wi;                                   // tensor_dim0_stride lo32 = 512
        g1[6] = 0;                                    // stride hi | dim1_stride lo
        g1[7] = 0;

        v4i gz4 = {0, 0, 0, 0};                       // 2D tensor: groups 2/3 unused
        v8i gz8 = {0, 0, 0, 0, 0, 0, 0, 0};
        __builtin_amdgcn_tensor_load_to_lds(g0, g1, gz4, gz4, gz8, 0);
    }
    __builtin_amdgcn_s_wait_tensorcnt(0);             // wait on TENSORcnt
    __syncthreads();

    // ---- 9-point stencil from LDS ----
    // out = 0.5 + 0.5*g - 0.25*(c0*(W+E) + c1*(NE+SW) + c2*(N+S) + c3*(NW+SE))
    const float c0 = 1.0f / 15.0f, c1 = 2.0f / 15.0f,
                c2 = 4.0f / 15.0f, c3 = 8.0f / 15.0f;
    float sum = 0.0f, sq = 0.0f;
    float* outrow = sout + (size_t)b * HW + (size_t)r0 * Wi;

    #pragma unroll 4
    for (int i = 0; i < (ROWS * Wi) / 256; ++i) {
        int p  = threadIdx.x + i * 256;
        int tr = p >> 9;                // tile row 0..15
        int w  = p & (Wi - 1);
        const float* c = &tile[(tr + 1) * Wi + w];
        float g  = c[0];
        float Nv = c[-Wi],  Sv = c[Wi];
        float Wv = (w > 0)      ? c[-1]      : 0.0f;
        float Ev = (w < Wi - 1) ? c[1]       : 0.0f;
        float NW = (w > 0)      ? c[-Wi - 1] : 0.0f;
        float NE = (w < Wi - 1) ? c[-Wi + 1] : 0.0f;
        float SW = (w > 0)      ? c[Wi - 1]  : 0.0f;
        float SE = (w < Wi - 1) ? c[Wi + 1]  : 0.0f;
        float acc = c0 * (Wv + Ev) + c1 * (NE + SW) + c2 * (Nv + Sv) + c3 * (NW + SE);
        float o = 0.5f + 0.5f * g - 0.25f * acc;
        sum += o;
        sq  += o * o;
        outrow[p] = o;   // RT policy: stays in L2 for kernel 4
    }

    // ---- deterministic in-LDS tree reduction ----
    red[threadIdx.x]       = sum;
    red[256 + threadIdx.x] = sq;
    __syncthreads();
    for (int s = 128; s > 0; s >>= 1) {
        if (threadIdx.x < s) {
            red[threadIdx.x]       += red[threadIdx.x + s];
            red[256 + threadIdx.x] += red[256 + threadIdx.x + s];
        }
        __syncthreads();
    }
    if (threadIdx.x == 0) {
        partial[blockIdx.x * 2]     = red[0];
        partial[blockIdx.x * 2 + 1] = red[256];
    }
}

// ---------------- Kernel 3: finalize per-image mean / inv-std ----------------
__global__ void stats_kernel(const float* __restrict__ partial, float* __restrict__ stats) {
    int b = threadIdx.x;
    if (b >= Bn) return;
    float sum = 0.0f, sq = 0.0f;
    for (int t = 0; t < 32; ++t) {          // fixed order -> deterministic
        sum += partial[(b * 32 + t) * 2];
        sq  += partial[(b * 32 + t) * 2 + 1];
    }
    float mean = sum * (1.0f / (float)HW);
    float var  = sq * (1.0f / (float)HW) - mean * mean;
    stats[b * 2]     = mean;
    stats[b * 2 + 1] = rsqrtf(var + 1e-5f);
}

// -------- Kernel 4: normalize + replicate to 3 channels (float4) --------
// s is last-use (NT load releases L2 lines); out is write-once never-reread
// (NT stores keep the 100 MB stream from occupying L2).
__global__ __launch_bounds__(256)
void norm_kernel(const float* __restrict__ s, const float* __restrict__ stats,
                 float* __restrict__ out, int n4) {
    int i = blockIdx.x * blockDim.x + threadIdx.x;
    if (i >= n4) return;
    long p   = (long)i * 4;
    int  b   = (int)(p >> 18);
    int  off = (int)(p & (HW - 1));
    v4f v = __builtin_nontemporal_load((const v4f*)(s + p));
    float mean = stats[2 * b];
    float inv  = stats[2 * b + 1];
    v4f r;
    r.x = (v.x - mean) * inv;
    r.y = (v.y - mean) * inv;
    r.z = (v.z - mean) * inv;
    r.w = (v.w - mean) * inv;
    float* base = out + (size_t)b * 3 * HW + off;
    __builtin_nontemporal_store(r, (v4f*)(base));
    __builtin_nontemporal_store(r, (v4f*)(base + HW));
    __builtin_nontemporal_store(r, (v4f*)(base + 2 * HW));
}

extern "C" void kernel_launch(void* const* d_in, const int* in_sizes, int n_in,
                              void* d_out, int out_size, void* d_ws, size_t ws_size,
                              hipStream_t stream) {
    const float* x = (const float*)d_in[0];
    float* out = (float*)d_out;

    char*  ws      = (char*)d_ws;
    float* gray    = (float*)ws;                                  // 33.5 MB
    float* s       = (float*)(ws + (size_t)Bn * HW * sizeof(float));   // 33.5 MB
    float* partial = (float*)(ws + (size_t)2 * Bn * HW * sizeof(float)); // 8 KB
    float* stats   = partial + Bn * 32 * 2;                       // 256 B

    const int n4 = Bn * HW / 4;   // 2,097,152 float4 groups

    gray_kernel   <<<n4 / 256, 256, 0, stream>>>(x, gray, n4);
    stencil_kernel<<<Bn * 32,  256, 0, stream>>>(gray, s, partial);
    stats_kernel  <<<1,        32,  0, stream>>>(partial, stats);
    norm_kernel   <<<n4 / 256, 256, 0, stream>>>(s, stats, out, n4);
}